// AdaptiveAggGCN_77249281786151
// MI455X (gfx1250) — compile-verified
//
#include <hip/hip_runtime.h>

// Problem constants (from reference)
constexpr int kN   = 30000;   // nodes
constexpr int kE   = 150000;  // edges per graph
constexpr int kHID = 512;     // IN == HID == 512
constexpr int kOUT = 256;
constexpr int kG   = 3;

constexpr int kKSLAB  = 16;                   // K-rows staged per TDM transfer
constexpr int kNSLAB  = kHID / kKSLAB;        // 32 slabs
constexpr int kSlabW  = kKSLAB * kHID;        // 8192 floats = 32 KB (W slab)
constexpr int kSlabWL = kKSLAB * kOUT;        // 4096 floats = 16 KB (W_lin slab)

typedef __attribute__((ext_vector_type(2))) float v2f;
typedef __attribute__((ext_vector_type(8))) float v8f;
typedef __attribute__((ext_vector_type(4))) unsigned int u32x4;
typedef __attribute__((ext_vector_type(4))) int i32x4;
typedef __attribute__((ext_vector_type(8))) int i32x8;

// ---- Tensor Data Mover availability / arity detection -----------------------
#if defined(__has_builtin)
#  if __has_builtin(__builtin_amdgcn_tensor_load_to_lds)
#    define HAVE_TDM 1
#  else
#    define HAVE_TDM 0
#  endif
#else
#  define HAVE_TDM 0
#endif
#if HAVE_TDM && __has_include(<hip/amd_detail/amd_gfx1250_TDM.h>)
#  define HAVE_TDM6 1   // therock clang-23 headers -> 6-arg builtin
#else
#  define HAVE_TDM6 0   // ROCm 7.2 clang-22 -> 5-arg builtin
#endif

// D(16x16,f32) = A(16x4,f32) * B(4x16,f32) + C
__device__ __forceinline__ v8f wmma_f32(v2f a, v2f b, v8f c) {
  return __builtin_amdgcn_wmma_f32_16x16x4_f32(
      /*neg_a=*/false, a, /*neg_b=*/false, b,
      /*c_mod=*/(short)0, c, /*reuse_a=*/false, /*reuse_b=*/false);
}

#if HAVE_TDM
// TDM load of a 2D f32 tile (kKSLAB rows x COLS cols, row stride COLS elements)
// from gsrc into LDS byte address lds_byte_addr. D# per ISA 08 §8.3-8.4.
template <int COLS>
__device__ __forceinline__ void tdm_load_slab(const float* gsrc, unsigned lds_byte_addr) {
  const unsigned long long ga = (unsigned long long)(const void*)gsrc;
  u32x4 g0;
  g0[0] = 1u;                                                  // count=1, user, no gather
  g0[1] = lds_byte_addr;                                       // lds_addr (bytes)
  g0[2] = (unsigned)ga;                                        // global_addr[31:0]
  g0[3] = (unsigned)((ga >> 32) & 0x01FFFFFFu) | (2u << 30);   // global_addr[56:32] | type=2
  i32x8 g1;
  g1[0] = 0x00020000;                                          // wg_mask=0, data_size=2 (4B)
  g1[1] = (int)(((unsigned)COLS & 0xFFFFu) << 16);             // tensor_dim0 lo16 @ [31:16]
  g1[2] = (int)((((unsigned)COLS >> 16) & 0xFFFFu)             // tensor_dim0 hi16
              | (((unsigned)kKSLAB & 0xFFFFu) << 16));         // tensor_dim1 lo16
  g1[3] = (int)(((unsigned)COLS & 0xFFFFu) << 16);             // tensor_dim1 hi=0 | tile_dim0
  g1[4] = kKSLAB;                                              // tile_dim1 | tile_dim2=0
  g1[5] = COLS;                                                // tensor_dim0_stride lo32
  g1[6] = 0;                                                   // stride0 hi | stride1 lo
  g1[7] = 0;
  i32x4 z4 = {0, 0, 0, 0};
#if HAVE_TDM6
  i32x8 z8 = {0, 0, 0, 0, 0, 0, 0, 0};
  __builtin_amdgcn_tensor_load_to_lds(g0, g1, z4, z4, z8, 0);
#else
  __builtin_amdgcn_tensor_load_to_lds(g0, g1, z4, z4, 0);
#endif
}
#endif  // HAVE_TDM

// ---------------------------------------------------------------------------
// H[M,512] = (X[M,512] * rs[m]) @ W[512,512]
// block = 512 thr = 16 waves; block tile = 64(M) x 512(N); wave tile = 32 x 64
// (two M-groups of 8 waves share each LDS slab). W staged via TDM, double-
// buffered 16x512 slabs (64 KB). grid.x = ceil(kN/64); A rows clamped so EXEC
// stays all-ones through every WMMA; stores bounds-checked afterwards.
// ---------------------------------------------------------------------------
__global__ __launch_bounds__(512, 1)
void gemm_xw_kernel(const float* __restrict__ X, const float* __restrict__ W,
                    const float* __restrict__ rs, float* __restrict__ H) {
  __shared__ float smem[2 * kSlabW];  // 64 KB

  const int lane = threadIdx.x & 31;
  const int wv   = threadIdx.x >> 5;   // 0..15
  const int half = lane >> 4;          // 0/1
  const int r    = lane & 15;
  const int mgrp = wv >> 3;            // 0/1
  const int n0   = (wv & 7) << 6;      // 64 cols per wave
  const int m0   = (blockIdx.x << 6) + (mgrp << 5);  // 32 rows per wave-group
  const int colb = n0 + r;

  const int row0 = m0 + r;
  const int row1 = m0 + 16 + r;
  const int cr0 = row0 < kN ? row0 : kN - 1;   // clamp (keep EXEC full)
  const int cr1 = row1 < kN ? row1 : kN - 1;
  const float s0 = rs[cr0];
  const float s1 = rs[cr1];
  const float* __restrict__ A0 = X + (size_t)cr0 * kHID;
  const float* __restrict__ A1 = X + (size_t)cr1 * kHID;

#if HAVE_TDM
  const unsigned lds_base = (unsigned)(unsigned long long)(void*)smem;
  if (wv == 0) {
    tdm_load_slab<kHID>(W, lds_base);            // preload slab 0 -> buf 0
    __builtin_amdgcn_s_wait_tensorcnt(0);
  }
  __syncthreads();
#else
  {  // cooperative preload of slab 0
    const float4* src = (const float4*)W;
    float4* dst = (float4*)smem;
    for (int i = threadIdx.x; i < kSlabW / 4; i += 512) dst[i] = src[i];
  }
  __syncthreads();
#endif

  v8f acc00 = {}, acc01 = {}, acc02 = {}, acc03 = {};
  v8f acc10 = {}, acc11 = {}, acc12 = {}, acc13 = {};

  for (int s = 0; s < kNSLAB; ++s) {
#if HAVE_TDM
    if (wv == 0 && s + 1 < kNSLAB)
      tdm_load_slab<kHID>(W + (size_t)(s + 1) * kSlabW,
                          lds_base + (unsigned)(((s + 1) & 1) * kSlabW * 4));
#else
    if (s + 1 < kNSLAB) {
      const float4* src = (const float4*)(W + (size_t)(s + 1) * kSlabW);
      float4* dst = (float4*)(smem + ((s + 1) & 1) * kSlabW);
      for (int i = threadIdx.x; i < kSlabW / 4; i += 512) dst[i] = src[i];
    }
#endif
    const float* __restrict__ Bs = smem + (s & 1) * kSlabW;
    const int sbase = s * kKSLAB;
#pragma unroll
    for (int kk = 0; kk < kKSLAB; kk += 4) {
      const int ka = sbase + kk + (half << 1);   // global K (A fragment)
      const int dk = kk + (half << 1);           // slab-local K (B fragment)
      const float2 av0 = *reinterpret_cast<const float2*>(A0 + ka);
      const float2 av1 = *reinterpret_cast<const float2*>(A1 + ka);
      v2f a0; a0[0] = av0.x * s0; a0[1] = av0.y * s0;
      v2f a1; a1[0] = av1.x * s1; a1[1] = av1.y * s1;

      const float* __restrict__ Bp = Bs + dk * kHID + colb;  // DS path
      v2f b0; b0[0] = Bp[0];  b0[1] = Bp[kHID + 0];
      v2f b1; b1[0] = Bp[16]; b1[1] = Bp[kHID + 16];
      v2f b2; b2[0] = Bp[32]; b2[1] = Bp[kHID + 32];
      v2f b3; b3[0] = Bp[48]; b3[1] = Bp[kHID + 48];

      acc00 = wmma_f32(a0, b0, acc00);
      acc10 = wmma_f32(a1, b0, acc10);
      acc01 = wmma_f32(a0, b1, acc01);
      acc11 = wmma_f32(a1, b1, acc11);
      acc02 = wmma_f32(a0, b2, acc02);
      acc12 = wmma_f32(a1, b2, acc12);
      acc03 = wmma_f32(a0, b3, acc03);
      acc13 = wmma_f32(a1, b3, acc13);
    }
#if HAVE_TDM
    if (wv == 0) __builtin_amdgcn_s_wait_tensorcnt(0);
#endif
    __syncthreads();
  }

  {
    const int rb0 = m0 + (half << 3);
#pragma unroll
    for (int v = 0; v < 8; ++v) {
      const int rr = rb0 + v;
      if (rr < kN) {
        float* __restrict__ Hp = H + (size_t)rr * kHID + colb;
        Hp[0] = acc00[v]; Hp[16] = acc01[v]; Hp[32] = acc02[v]; Hp[48] = acc03[v];
      }
    }
    const int rb1 = m0 + 16 + (half << 3);
#pragma unroll
    for (int v = 0; v < 8; ++v) {
      const int rr = rb1 + v;
      if (rr < kN) {
        float* __restrict__ Hp = H + (size_t)rr * kHID + colb;
        Hp[0] = acc10[v]; Hp[16] = acc11[v]; Hp[32] = acc12[v]; Hp[48] = acc13[v];
      }
    }
  }
}

// ---------------------------------------------------------------------------
// OUT[M,256] (+)= T[M,512] @ WL[512,256]   (init: acc=b_lin ; else acc=OUT)
// block = 256 thr = 8 waves; block tile = 64(M) x 256(N); wave tile = 32 x 64
// (two M-groups of 4 waves). WL staged via TDM, double-buffered 16x256 slabs
// (32 KB). grid.x = ceil(kN/64).
// ---------------------------------------------------------------------------
__global__ __launch_bounds__(256, 2)
void gemm_out_kernel(const float* __restrict__ T, const float* __restrict__ WL,
                     const float* __restrict__ bias, float* __restrict__ OUT,
                     int init) {
  __shared__ float smem[2 * kSlabWL];  // 32 KB

  const int lane = threadIdx.x & 31;
  const int wv   = threadIdx.x >> 5;   // 0..7
  const int half = lane >> 4;
  const int r    = lane & 15;
  const int mgrp = wv >> 2;            // 0/1
  const int n0   = (wv & 3) << 6;      // 64 cols per wave
  const int m0   = (blockIdx.x << 6) + (mgrp << 5);
  const int colb = n0 + r;

  const int row0 = m0 + r;
  const int row1 = m0 + 16 + r;
  const int cr0 = row0 < kN ? row0 : kN - 1;
  const int cr1 = row1 < kN ? row1 : kN - 1;
  const float* __restrict__ A0 = T + (size_t)cr0 * kHID;
  const float* __restrict__ A1 = T + (size_t)cr1 * kHID;

#if HAVE_TDM
  const unsigned lds_base = (unsigned)(unsigned long long)(void*)smem;
  if (wv == 0) {
    tdm_load_slab<kOUT>(WL, lds_base);
    __builtin_amdgcn_s_wait_tensorcnt(0);
  }
  __syncthreads();
#else
  {
    const float4* src = (const float4*)WL;
    float4* dst = (float4*)smem;
    for (int i = threadIdx.x; i < kSlabWL / 4; i += 256) dst[i] = src[i];
  }
  __syncthreads();
#endif

  const int rb0 = m0 + (half << 3);
  const int rb1 = m0 + 16 + (half << 3);

  v8f acc00, acc01, acc02, acc03, acc10, acc11, acc12, acc13;
  if (init) {
    const float c0 = bias[colb + 0],  c1 = bias[colb + 16];
    const float c2 = bias[colb + 32], c3 = bias[colb + 48];
#pragma unroll
    for (int v = 0; v < 8; ++v) {
      acc00[v] = c0; acc01[v] = c1; acc02[v] = c2; acc03[v] = c3;
      acc10[v] = c0; acc11[v] = c1; acc12[v] = c2; acc13[v] = c3;
    }
  } else {
#pragma unroll
    for (int v = 0; v < 8; ++v) {
      const int r0 = rb0 + v; const int c0r = r0 < kN ? r0 : kN - 1;
      const float* __restrict__ Op0 = OUT + (size_t)c0r * kOUT + colb;
      acc00[v] = Op0[0]; acc01[v] = Op0[16]; acc02[v] = Op0[32]; acc03[v] = Op0[48];
      const int r1 = rb1 + v; const int c1r = r1 < kN ? r1 : kN - 1;
      const float* __restrict__ Op1 = OUT + (size_t)c1r * kOUT + colb;
      acc10[v] = Op1[0]; acc11[v] = Op1[16]; acc12[v] = Op1[32]; acc13[v] = Op1[48];
    }
  }

  for (int s = 0; s < kNSLAB; ++s) {
#if HAVE_TDM
    if (wv == 0 && s + 1 < kNSLAB)
      tdm_load_slab<kOUT>(WL + (size_t)(s + 1) * kSlabWL,
                          lds_base + (unsigned)(((s + 1) & 1) * kSlabWL * 4));
#else
    if (s + 1 < kNSLAB) {
      const float4* src = (const float4*)(WL + (size_t)(s + 1) * kSlabWL);
      float4* dst = (float4*)(smem + ((s + 1) & 1) * kSlabWL);
      for (int i = threadIdx.x; i < kSlabWL / 4; i += 256) dst[i] = src[i];
    }
#endif
    const float* __restrict__ Bs = smem + (s & 1) * kSlabWL;
    const int sbase = s * kKSLAB;
#pragma unroll
    for (int kk = 0; kk < kKSLAB; kk += 4) {
      const int ka = sbase + kk + (half << 1);
      const int dk = kk + (half << 1);
      const float2 av0 = *reinterpret_cast<const float2*>(A0 + ka);
      const float2 av1 = *reinterpret_cast<const float2*>(A1 + ka);
      v2f a0; a0[0] = av0.x; a0[1] = av0.y;
      v2f a1; a1[0] = av1.x; a1[1] = av1.y;

      const float* __restrict__ Bp = Bs + dk * kOUT + colb;
      v2f b0; b0[0] = Bp[0];  b0[1] = Bp[kOUT + 0];
      v2f b1; b1[0] = Bp[16]; b1[1] = Bp[kOUT + 16];
      v2f b2; b2[0] = Bp[32]; b2[1] = Bp[kOUT + 32];
      v2f b3; b3[0] = Bp[48]; b3[1] = Bp[kOUT + 48];

      acc00 = wmma_f32(a0, b0, acc00);
      acc10 = wmma_f32(a1, b0, acc10);
      acc01 = wmma_f32(a0, b1, acc01);
      acc11 = wmma_f32(a1, b1, acc11);
      acc02 = wmma_f32(a0, b2, acc02);
      acc12 = wmma_f32(a1, b2, acc12);
      acc03 = wmma_f32(a0, b3, acc03);
      acc13 = wmma_f32(a1, b3, acc13);
    }
#if HAVE_TDM
    if (wv == 0) __builtin_amdgcn_s_wait_tensorcnt(0);
#endif
    __syncthreads();
  }

#pragma unroll
  for (int v = 0; v < 8; ++v) {
    const int r0 = rb0 + v;
    if (r0 < kN) {
      float* __restrict__ Op = OUT + (size_t)r0 * kOUT + colb;
      Op[0] = acc00[v]; Op[16] = acc01[v]; Op[32] = acc02[v]; Op[48] = acc03[v];
    }
    const int r1 = rb1 + v;
    if (r1 < kN) {
      float* __restrict__ Op = OUT + (size_t)r1 * kOUT + colb;
      Op[0] = acc10[v]; Op[16] = acc11[v]; Op[32] = acc12[v]; Op[48] = acc13[v];
    }
  }
}

// ---------------------------------------------------------------------------
// Degrees (exact, integer atomics). cnt layout: [g][2][kN] ints.
// ---------------------------------------------------------------------------
__global__ void degree_kernel(const int* __restrict__ e0, const int* __restrict__ e1,
                              const int* __restrict__ e2, int* __restrict__ cnt) {
  const int i = blockIdx.x * blockDim.x + threadIdx.x;
  if (i >= kG * kE) return;
  const int g = i / kE;
  const int e = i - g * kE;
  const int* ed = (g == 0) ? e0 : (g == 1) ? e1 : e2;
  atomicAdd(cnt + (size_t)(2 * g + 0) * kN + ed[e],      1);  // out-deg (src)
  atomicAdd(cnt + (size_t)(2 * g + 1) * kN + ed[kE + e], 1);  // in-deg  (dst)
}

__global__ void rsqrt_kernel(const int* __restrict__ cnt,
                             float* __restrict__ rs_out, float* __restrict__ rs_in) {
  const int i = blockIdx.x * blockDim.x + threadIdx.x;
  if (i >= kG * kN) return;
  const int g = i / kN;
  const int n = i - g * kN;
  const int co = cnt[(size_t)(2 * g + 0) * kN + n];
  const int ci = cnt[(size_t)(2 * g + 1) * kN + n];
  rs_out[i] = rsqrtf((float)(co > 1 ? co : 1));
  rs_in[i]  = rsqrtf((float)(ci > 1 ? ci : 1));
}

// ---------------------------------------------------------------------------
// Scatter: T[dst] += H[src]  (one block per edge, 512 feats / 256 thr)
// ---------------------------------------------------------------------------
__global__ __launch_bounds__(256)
void scatter_kernel(const int* __restrict__ src, const int* __restrict__ dst,
                    const float* __restrict__ H, float* __restrict__ T) {
  const int e = blockIdx.x;
  const int s = src[e];
  const int d = dst[e];
  const float* __restrict__ hp = H + (size_t)s * kHID;
  float* __restrict__ tp = T + (size_t)d * kHID;
  for (int f = threadIdx.x; f < kHID; f += blockDim.x)
    unsafeAtomicAdd(tp + f, hp[f]);
}

// T = tanh(softmax(alphas)[g] * (T * rs_in[n] + bias[f]))
__global__ void post_kernel(float* __restrict__ T, const float* __restrict__ rsin,
                            const float* __restrict__ bias,
                            const float* __restrict__ alphas, int g) {
  const long long i = (long long)blockIdx.x * blockDim.x + threadIdx.x;
  if (i >= (long long)kN * kHID) return;
  const int n = (int)(i >> 9);
  const int f = (int)(i & (kHID - 1));
  const float a0 = alphas[0], a1 = alphas[1], a2 = alphas[2];
  const float m = fmaxf(a0, fmaxf(a1, a2));
  const float x0 = __expf(a0 - m), x1 = __expf(a1 - m), x2 = __expf(a2 - m);
  const float ag = ((g == 0) ? x0 : (g == 1) ? x1 : x2) / (x0 + x1 + x2);
  T[i] = tanhf(ag * (T[i] * rsin[n] + bias[f]));
}

__global__ void zero_f32_kernel(float* __restrict__ p, long long n) {
  const long long i = (long long)blockIdx.x * blockDim.x + threadIdx.x;
  if (i < n) p[i] = 0.0f;
}
__global__ void zero_i32_kernel(int* __restrict__ p, long long n) {
  const long long i = (long long)blockIdx.x * blockDim.x + threadIdx.x;
  if (i < n) p[i] = 0;
}

// ---------------------------------------------------------------------------
// d_in order (setup_inputs dict order):
// 0:x 1:edges0 2:W0 3:b0 4:edges1 5:W1 6:b1 7:edges2 8:W2 9:b2 10:alphas 11:W_lin 12:b_lin
// ---------------------------------------------------------------------------
extern "C" void kernel_launch(void* const* d_in, const int* in_sizes, int n_in,
                              void* d_out, int out_size, void* d_ws, size_t ws_size,
                              hipStream_t stream) {
  const float* x      = (const float*)d_in[0];
  const float* alphas = (const float*)d_in[10];
  const float* W_lin  = (const float*)d_in[11];
  const float* b_lin  = (const float*)d_in[12];
  const int*   edges[kG] = { (const int*)d_in[1], (const int*)d_in[4], (const int*)d_in[7] };
  const float* W[kG]     = { (const float*)d_in[2], (const float*)d_in[5], (const float*)d_in[8] };
  const float* b[kG]     = { (const float*)d_in[3], (const float*)d_in[6], (const float*)d_in[9] };
  float* out = (float*)d_out;

  // Workspace carving (256B aligned)
  auto align_up = [](size_t v) { return (v + 255) & ~(size_t)255; };
  char* base = (char*)d_ws;
  size_t off = 0;
  float* rs_out = (float*)(base + off); off = align_up(off + sizeof(float) * kG * kN);
  float* rs_in  = (float*)(base + off); off = align_up(off + sizeof(float) * kG * kN);
  int*   cnt    = (int*)  (base + off); off = align_up(off + sizeof(int) * 2 * kG * kN);
  float* h      = (float*)(base + off); off = align_up(off + sizeof(float) * (size_t)kN * kHID);
  float* t      = (float*)(base + off); off = align_up(off + sizeof(float) * (size_t)kN * kHID);
  (void)ws_size; (void)in_sizes; (void)n_in; (void)out_size;

  const long long nh = (long long)kN * kHID;
  const int gridM64 = (kN + 63) / 64;  // 469

  // Degrees -> rsqrt norms (all 3 graphs)
  zero_i32_kernel<<<(2 * kG * kN + 255) / 256, 256, 0, stream>>>(cnt, (long long)2 * kG * kN);
  degree_kernel<<<(kG * kE + 255) / 256, 256, 0, stream>>>(edges[0], edges[1], edges[2], cnt);
  rsqrt_kernel<<<(kG * kN + 255) / 256, 256, 0, stream>>>(cnt, rs_out, rs_in);

  for (int g = 0; g < kG; ++g) {
    // h = (x * rs_out_g) @ W_g   (TDM-staged W, WMMA f32)
    gemm_xw_kernel<<<gridM64, 512, 0, stream>>>(x, W[g], rs_out + (size_t)g * kN, h);
    // t = 0 ; t[dst] += h[src]
    zero_f32_kernel<<<(int)((nh + 255) / 256), 256, 0, stream>>>(t, nh);
    scatter_kernel<<<kE, 256, 0, stream>>>(edges[g], edges[g] + kE, h, t);
    // t = tanh(a_g * (t * rs_in_g + b_g))
    post_kernel<<<(int)((nh + 255) / 256), 256, 0, stream>>>(t, rs_in + (size_t)g * kN, b[g], alphas, g);
    // out (+)= t @ W_lin[g*512 : (g+1)*512, :]  (+ b_lin on first branch)
    gemm_out_kernel<<<gridM64, 256, 0, stream>>>(t, W_lin + (size_t)g * kHID * kOUT,
                                                 b_lin, out, g == 0 ? 1 : 0);
  }
}